// Projection_68650757259663
// MI455X (gfx1250) — compile-verified
//
#include <hip/hip_runtime.h>

typedef float v2f __attribute__((ext_vector_type(2)));
typedef float v4f __attribute__((ext_vector_type(4)));
typedef float v8f __attribute__((ext_vector_type(8)));

#define N_NODES 100000
#define EMBED   256
#define N_EDGES 1600000

// -------- GEMM: proj[M,256] = x[M,256] @ W^T + b, via V_WMMA_F32_16X16X4_F32 --------
// Block: 256 threads (8 waves). Block tile: 128(M) x 64(N). Wave tile: 16(M) x 64(N)
// as four 16x16 accumulators. K chunked by 64 and staged in LDS (padded stride 68
// floats -> conflict-free float2 fragment reads on 64 banks).
__global__ __launch_bounds__(256) void proj_wmma_kernel(
    const float* __restrict__ x, const float* __restrict__ W,
    const float* __restrict__ bias, float* __restrict__ proj)
{
    __shared__ float Xs[128][68];
    __shared__ float Ws[64][68];

    const int tid   = threadIdx.x;
    const int wave  = tid >> 5;
    const int lane  = tid & 31;
    const int half  = lane >> 4;   // 0: lanes 0-15, 1: lanes 16-31
    const int l16   = lane & 15;
    const int mBase = blockIdx.x * 128;
    const int nBase = blockIdx.y * 64;

    v8f acc[4] = {};

    for (int kc = 0; kc < 4; ++kc) {           // K chunks of 64
        __syncthreads();                        // protect LDS from previous chunk readers
        // Stage x tile: 128 rows x 64 cols, coalesced float4
        #pragma unroll
        for (int i = 0; i < 8; ++i) {
            int flat = tid + i * 256;           // float4 index over [128][16]
            int row  = flat >> 4;
            int c4   = flat & 15;
            int gm   = mBase + row;
            if (gm >= N_NODES) gm = N_NODES - 1;          // clamp OOB reads
            v4f v = *(const v4f*)(x + (size_t)gm * EMBED + kc * 64 + c4 * 4);
            *(v4f*)(&Xs[row][c4 * 4]) = v;
        }
        // Stage W tile: 64 rows (out-features) x 64 cols (in-features)
        #pragma unroll
        for (int i = 0; i < 4; ++i) {
            int flat = tid + i * 256;
            int row  = flat >> 4;
            int c4   = flat & 15;
            v4f v = *(const v4f*)(W + (size_t)(nBase + row) * EMBED + kc * 64 + c4 * 4);
            *(v4f*)(&Ws[row][c4 * 4]) = v;
        }
        __syncthreads();

        // 16 K-steps of 4, 4 WMMAs each (A fragment reused across the 4 N-tiles)
        for (int kk = 0; kk < 64; kk += 4) {
            // A 16x4 f32 layout: lanes 0-15 -> M=lane, {K0,K1}; lanes 16-31 -> {K2,K3}
            v2f a = *(const v2f*)(&Xs[wave * 16 + l16][kk + 2 * half]);
            #pragma unroll
            for (int t = 0; t < 4; ++t) {
                // B 4x16 layout mirrored: lanes 0-15 -> N=lane, {K0,K1}; lanes 16-31 -> {K2,K3}
                v2f bf = *(const v2f*)(&Ws[t * 16 + l16][kk + 2 * half]);
                acc[t] = __builtin_amdgcn_wmma_f32_16x16x4_f32(
                    /*neg_a=*/false, a, /*neg_b=*/false, bf,
                    /*c_mod=*/(short)0, acc[t],
                    /*reuse_a=*/false, /*reuse_b=*/false);
            }
        }
    }

    // Epilogue: bias add + store. C/D layout: lane -> N = l16; VGPR r -> M = r + 8*half.
    #pragma unroll
    for (int t = 0; t < 4; ++t) {
        int n   = nBase + t * 16 + l16;
        float bv = bias[n];
        int m0  = mBase + wave * 16 + 8 * half;
        #pragma unroll
        for (int r = 0; r < 8; ++r) {
            int m = m0 + r;
            if (m < N_NODES)
                proj[(size_t)m * EMBED + n] = acc[t][r] + bv;
        }
    }
}

// -------- Gather: out[e,:] = proj[index[e],:]. 4 edges / 256-thread block, ----------
// 128-bit accesses; proj reads stay regular (L2-resident, proj=102MB < 192MB L2),
// output stream (1.64 GB, never re-read) uses non-temporal stores.
__global__ __launch_bounds__(256) void gather_kernel(
    const float* __restrict__ proj, const int* __restrict__ index,
    float* __restrict__ out)
{
    int e = blockIdx.x * 4 + (threadIdx.x >> 6);   // 64 threads per edge
    int j = threadIdx.x & 63;                      // float4 slot within the 256-f row
    int node = index[e];
    v4f v = ((const v4f*)(proj + (size_t)node * EMBED))[j];
    __builtin_nontemporal_store(v, (v4f*)(out + (size_t)e * EMBED) + j);
}

extern "C" void kernel_launch(void* const* d_in, const int* in_sizes, int n_in,
                              void* d_out, int out_size, void* d_ws, size_t ws_size,
                              hipStream_t stream) {
    const float* x     = (const float*)d_in[0];
    const int*   index = (const int*)  d_in[1];
    const float* W     = (const float*)d_in[2];
    const float* bias  = (const float*)d_in[3];
    float* proj = (float*)d_ws;          // 100000*256*4 = 102.4 MB scratch
    float* out  = (float*)d_out;

    dim3 gridGemm((N_NODES + 127) / 128, EMBED / 64);   // 782 x 4 blocks
    proj_wmma_kernel<<<gridGemm, 256, 0, stream>>>(x, W, bias, proj);

    gather_kernel<<<N_EDGES / 4, 256, 0, stream>>>(proj, index, out);
}